// SlicedBlockFaster_76914274337000
// MI455X (gfx1250) — compile-verified
//
#include <hip/hip_runtime.h>
#include <math.h>

// ---------------- problem constants ----------------
#define HH    112
#define SW    7
#define N49   49
#define DIMC  192
#define HEADS 6
#define HD    32
#define QKVD  576
#define HID   768
#define NWIN  256   // 16x16 windows per image
#define DSH   3     // disturb shift

typedef float v2f __attribute__((ext_vector_type(2)));
typedef float v8f __attribute__((ext_vector_type(8)));

// CDNA5 fp32 WMMA: D(16x16) = A(16x4) * B(4x16) + C
__device__ __forceinline__ v8f wmma4(v2f a, v2f b, v8f c) {
  return __builtin_amdgcn_wmma_f32_16x16x4_f32(false, a, false, b, (short)0, c,
                                               false, false);
}

__device__ __forceinline__ float wave_red_sum(float v) {
#pragma unroll
  for (int o = 16; o > 0; o >>= 1) v += __shfl_xor(v, o, 32);
  return v;
}

// shift-mask band of a coordinate in the rolled image: [0,105)=0, [105,109)=1, [109,112)=2
__device__ __forceinline__ int band3(int v) { return v < 105 ? 0 : (v < 109 ? 1 : 2); }

// LDS geometry for kernel 1 (all row dims padded to 64 so D-tile stores are
// unconditional; pad rows of s_xo are zeroed -> all downstream pads finite)
#define XS 200  // s_xo stride (floats)
#define SS 56   // score stride
#define QS 580  // qkv stride
#define MROWS 64

// =====================================================================
// Kernel 1: LN1 + shifted window partition + RPE/masked attention +
//           proj + residual scatter (one workgroup per 49-token window)
// =====================================================================
__global__ __launch_bounds__(256) void k_win_attn(
    const float* __restrict__ x, const float* __restrict__ w_qkv,
    const float* __restrict__ b_qkv, const float* __restrict__ w_proj,
    const float* __restrict__ b_proj, const float* __restrict__ rpe,
    const float* __restrict__ g1, const float* __restrict__ be1,
    float* __restrict__ out) {
  extern __shared__ float smem[];
  float* s_xo  = smem;                         // 64 x 200 : LN'd window, later attn out
  float* s_sc  = smem + MROWS * XS;            // 6 x 64 x 56 : per-head scores
  float* s_qkv = s_sc + HEADS * MROWS * SS;    // 64 x 580 : qkv

  const int tid   = threadIdx.x;
  const int lane  = tid & 31;
  const int wave  = tid >> 5;
  const int colid = lane & 15;
  const int khi   = lane >> 4;  // 0/1 : which K pair this lane holds
  const int kof   = khi * 2;

  const int wblk = blockIdx.x;
  const int bimg = wblk / NWIN;
  const int widx = wblk % NWIN;
  const int wy = widx >> 4, wx = widx & 15;

  // zero the pad rows of s_xo (rows 49..63) so pad GEMM outputs stay finite
  for (int idx = tid; idx < (MROWS - N49) * XS; idx += 256)
    s_xo[N49 * XS + idx] = 0.0f;

  // ---- LayerNorm1 + shifted-window gather (img = roll(x,-3,-3)) ----
  for (int t = wave; t < N49; t += 8) {
    int i = t / 7, j = t % 7;
    int oh = wy * 7 + i + DSH; if (oh >= HH) oh -= HH;
    int ow = wx * 7 + j + DSH; if (ow >= HH) ow -= HH;
    const float* xr = x + ((size_t)bimg * HH * HH + (size_t)oh * HH + ow) * DIMC;
    float v0 = xr[lane], v1 = xr[lane + 32], v2 = xr[lane + 64];
    float v3 = xr[lane + 96], v4 = xr[lane + 128], v5 = xr[lane + 160];
    float mu = wave_red_sum(v0 + v1 + v2 + v3 + v4 + v5) * (1.0f / 192.0f);
    float d0 = v0 - mu, d1 = v1 - mu, d2 = v2 - mu;
    float d3 = v3 - mu, d4 = v4 - mu, d5 = v5 - mu;
    float var = wave_red_sum(d0 * d0 + d1 * d1 + d2 * d2 + d3 * d3 + d4 * d4 + d5 * d5)
                * (1.0f / 192.0f);
    float rstd = rsqrtf(var + 1e-5f);
    float* dst = s_xo + t * XS;
    dst[lane]       = d0 * rstd * g1[lane]       + be1[lane];
    dst[lane + 32]  = d1 * rstd * g1[lane + 32]  + be1[lane + 32];
    dst[lane + 64]  = d2 * rstd * g1[lane + 64]  + be1[lane + 64];
    dst[lane + 96]  = d3 * rstd * g1[lane + 96]  + be1[lane + 96];
    dst[lane + 128] = d4 * rstd * g1[lane + 128] + be1[lane + 128];
    dst[lane + 160] = d5 * rstd * g1[lane + 160] + be1[lane + 160];
  }
  __syncthreads();

  // ---- QKV GEMM: s_qkv[64x576] = s_xo(64x192) @ w_qkv^T + b_qkv ----
  for (int tile = wave; tile < 4 * 36; tile += 8) {
    int m0 = (tile & 3) * 16, n0 = (tile >> 2) * 16;
    float bv = b_qkv[n0 + colid];
    v8f acc = {bv, bv, bv, bv, bv, bv, bv, bv};
    const float* ab = s_xo + (m0 + colid) * XS + kof;
    const float* wb = w_qkv + (size_t)(n0 + colid) * DIMC + kof;
#pragma unroll 8
    for (int k = 0; k < DIMC; k += 4) {
      v2f a = *(const v2f*)(ab + k);
      v2f b = *(const v2f*)(wb + k);
      acc = wmma4(a, b, acc);
    }
    float* dp = s_qkv + (m0 + 8 * khi) * QS + n0 + colid;
#pragma unroll
    for (int v = 0; v < 8; ++v) dp[v * QS] = acc[v];
  }
  __syncthreads();

  // ---- per-head attention: one head per wave ----
  if (wave < HEADS) {
    const int h = wave;
    float* sc = s_sc + h * MROWS * SS;
    const int qo = h * HD, ko2 = DIMC + h * HD, vo = 2 * DIMC + h * HD;

    // scores = q @ k^T  (49x32 @ 32x49, M/N padded to 64)
    for (int mt = 0; mt < 4; ++mt)
      for (int nt = 0; nt < 4; ++nt) {
        v8f acc = {};
        const float* ab = s_qkv + (mt * 16 + colid) * QS + qo + kof;
        const float* bb = s_qkv + (nt * 16 + colid) * QS + ko2 + kof;
#pragma unroll
        for (int k = 0; k < HD; k += 4) {
          v2f a = *(const v2f*)(ab + k);
          v2f b = *(const v2f*)(bb + k);
          acc = wmma4(a, b, acc);
        }
        float* dp = sc + (mt * 16 + 8 * khi) * SS + nt * 16 + colid;
#pragma unroll
        for (int v = 0; v < 8; ++v) dp[v * SS] = acc[v];
      }

    // fused scale + relative-position bias + shift mask + softmax (row per lane)
    const float scale = 0.17677669529663687f;  // 32^-0.5
    for (int r = lane; r < N49; r += 32) {
      int i1 = r / 7, j1 = r % 7;
      int cat1 = band3(wy * 7 + i1) * 3 + band3(wx * 7 + j1);
      float* sp = sc + r * SS;
      float mx = -1e30f;
      for (int c = 0; c < N49; ++c) {
        int i2 = c / 7, j2 = c % 7;
        int cat2 = band3(wy * 7 + i2) * 3 + band3(wx * 7 + j2);
        int ridx = (i1 - i2 + 6) * 13 + (j1 - j2 + 6);
        float val = sp[c] * scale + rpe[ridx * HEADS + h];
        if (cat1 != cat2) val -= 100.0f;  // shift mask
        sp[c] = val;
        mx = fmaxf(mx, val);
      }
      float sum = 0.0f;
      for (int c = 0; c < N49; ++c) { float e = expf(sp[c] - mx); sp[c] = e; sum += e; }
      float inv = 1.0f / sum;
      for (int c = 0; c < N49; ++c) sp[c] *= inv;
      for (int c = N49; c < SS; ++c) sp[c] = 0.0f;  // zero K padding
    }

    // out_h = attn @ v  (49x49 @ 49x32, K padded to 52 with zero attn cols)
    for (int mt = 0; mt < 4; ++mt)
      for (int nt = 0; nt < 2; ++nt) {
        v8f acc = {};
        const float* ab = sc + (mt * 16 + colid) * SS + kof;
#pragma unroll
        for (int k = 0; k < 52; k += 4) {
          v2f a = *(const v2f*)(ab + k);
          v2f b;
          b.x = s_qkv[(k + kof) * QS + vo + nt * 16 + colid];
          b.y = s_qkv[(k + kof + 1) * QS + vo + nt * 16 + colid];
          acc = wmma4(a, b, acc);
        }
        float* dp = s_xo + (mt * 16 + 8 * khi) * XS + h * HD + nt * 16 + colid;
#pragma unroll
        for (int v = 0; v < 8; ++v) dp[v * XS] = acc[v];
      }
  }
  __syncthreads();

  // ---- proj GEMM + residual scatter-add (inverse shift folded into index) ----
  for (int tile = wave; tile < 4 * 12; tile += 8) {
    int m0 = (tile & 3) * 16, n0 = (tile >> 2) * 16;
    float bv = b_proj[n0 + colid];
    v8f acc = {bv, bv, bv, bv, bv, bv, bv, bv};
    const float* ab = s_xo + (m0 + colid) * XS + kof;
    const float* wb = w_proj + (size_t)(n0 + colid) * DIMC + kof;
#pragma unroll 8
    for (int k = 0; k < DIMC; k += 4) {
      v2f a = *(const v2f*)(ab + k);
      v2f b = *(const v2f*)(wb + k);
      acc = wmma4(a, b, acc);
    }
#pragma unroll
    for (int v = 0; v < 8; ++v) {
      int r = m0 + v + 8 * khi;
      if (r < N49) {  // global scatter must stay inside the window
        int i = r / 7, j = r % 7;
        int oh = wy * 7 + i + DSH; if (oh >= HH) oh -= HH;
        int ow = wx * 7 + j + DSH; if (ow >= HH) ow -= HH;
        size_t gi = ((size_t)bimg * HH * HH + (size_t)oh * HH + ow) * DIMC + n0 + colid;
        out[gi] = x[gi] + acc[v];  // x1 = x + attn_out
      }
    }
  }
}

// =====================================================================
// Kernel 2: LN2 + fc1 + exact GELU + fc2 + residual (32 tokens / block)
// operates in place on d_out (holds x1 from kernel 1)
// =====================================================================
__global__ __launch_bounds__(256) void k_mlp(
    const float* __restrict__ g2, const float* __restrict__ be2,
    const float* __restrict__ w1, const float* __restrict__ b1,
    const float* __restrict__ w2, const float* __restrict__ b2,
    float* __restrict__ io) {
  extern __shared__ float smem[];
  const int YS = 200, HS = 776;
  float* s_y = smem;            // 32 x 200 : LN'd tokens
  float* s_h = smem + 32 * YS;  // 32 x 776 : gelu(fc1)

  const int tid = threadIdx.x, lane = tid & 31, wave = tid >> 5;
  const int colid = lane & 15, khi = lane >> 4, kof = khi * 2;
  const size_t T0 = (size_t)blockIdx.x * 32;

  // ---- LayerNorm2 ----
  for (int t = wave; t < 32; t += 8) {
    const float* xr = io + (T0 + t) * DIMC;
    float v0 = xr[lane], v1 = xr[lane + 32], v2 = xr[lane + 64];
    float v3 = xr[lane + 96], v4 = xr[lane + 128], v5 = xr[lane + 160];
    float mu = wave_red_sum(v0 + v1 + v2 + v3 + v4 + v5) * (1.0f / 192.0f);
    float d0 = v0 - mu, d1 = v1 - mu, d2 = v2 - mu;
    float d3 = v3 - mu, d4 = v4 - mu, d5 = v5 - mu;
    float var = wave_red_sum(d0 * d0 + d1 * d1 + d2 * d2 + d3 * d3 + d4 * d4 + d5 * d5)
                * (1.0f / 192.0f);
    float rstd = rsqrtf(var + 1e-5f);
    float* dst = s_y + t * YS;
    dst[lane]       = d0 * rstd * g2[lane]       + be2[lane];
    dst[lane + 32]  = d1 * rstd * g2[lane + 32]  + be2[lane + 32];
    dst[lane + 64]  = d2 * rstd * g2[lane + 64]  + be2[lane + 64];
    dst[lane + 96]  = d3 * rstd * g2[lane + 96]  + be2[lane + 96];
    dst[lane + 128] = d4 * rstd * g2[lane + 128] + be2[lane + 128];
    dst[lane + 160] = d5 * rstd * g2[lane + 160] + be2[lane + 160];
  }
  __syncthreads();

  // ---- fc1 (32x192 @ 192x768) + exact GELU ----
  for (int tile = wave; tile < 2 * 48; tile += 8) {
    int m0 = (tile & 1) * 16, n0 = (tile >> 1) * 16;
    float bv = b1[n0 + colid];
    v8f acc = {bv, bv, bv, bv, bv, bv, bv, bv};
    const float* ab = s_y + (m0 + colid) * YS + kof;
    const float* wb = w1 + (size_t)(n0 + colid) * DIMC + kof;
#pragma unroll 8
    for (int k = 0; k < DIMC; k += 4) {
      v2f a = *(const v2f*)(ab + k);
      v2f b = *(const v2f*)(wb + k);
      acc = wmma4(a, b, acc);
    }
    float* dp = s_h + (m0 + 8 * khi) * HS + n0 + colid;
#pragma unroll
    for (int v = 0; v < 8; ++v) {
      float u = acc[v];
      dp[v * HS] = 0.5f * u * (1.0f + erff(u * 0.7071067811865476f));
    }
  }
  __syncthreads();

  // ---- fc2 (32x768 @ 768x192) + residual ----
  for (int tile = wave; tile < 2 * 12; tile += 8) {
    int m0 = (tile & 1) * 16, n0 = (tile >> 1) * 16;
    float bv = b2[n0 + colid];
    v8f acc = {bv, bv, bv, bv, bv, bv, bv, bv};
    const float* ab = s_h + (m0 + colid) * HS + kof;
    const float* wb = w2 + (size_t)(n0 + colid) * HID + kof;
#pragma unroll 8
    for (int k = 0; k < HID; k += 4) {
      v2f a = *(const v2f*)(ab + k);
      v2f b = *(const v2f*)(wb + k);
      acc = wmma4(a, b, acc);
    }
#pragma unroll
    for (int v = 0; v < 8; ++v) {
      int r = m0 + v + 8 * khi;
      size_t gi = (T0 + r) * DIMC + n0 + colid;
      io[gi] = io[gi] + acc[v];  // out = x1 + mlp(ln(x1))
    }
  }
}

// =====================================================================
extern "C" void kernel_launch(void* const* d_in, const int* in_sizes, int n_in,
                              void* d_out, int out_size, void* d_ws, size_t ws_size,
                              hipStream_t stream) {
  (void)in_sizes; (void)n_in; (void)d_ws; (void)ws_size; (void)out_size;
  const float* x      = (const float*)d_in[0];
  const float* w_qkv  = (const float*)d_in[1];
  const float* b_qkv  = (const float*)d_in[2];
  const float* w_proj = (const float*)d_in[3];
  const float* b_proj = (const float*)d_in[4];
  const float* rpe    = (const float*)d_in[5];
  const float* g1     = (const float*)d_in[6];
  const float* be1    = (const float*)d_in[7];
  const float* g2     = (const float*)d_in[8];
  const float* be2    = (const float*)d_in[9];
  const float* w1     = (const float*)d_in[10];
  const float* b1     = (const float*)d_in[11];
  const float* w2     = (const float*)d_in[12];
  const float* b2     = (const float*)d_in[13];
  float* out = (float*)d_out;

  // Kernel 1: one workgroup per window (16 imgs * 256 windows)
  size_t lds1 = (size_t)(MROWS * XS + HEADS * MROWS * SS + MROWS * QS) * sizeof(float);
  k_win_attn<<<16 * NWIN, 256, lds1, stream>>>(x, w_qkv, b_qkv, w_proj, b_proj,
                                               rpe, g1, be1, out);

  // Kernel 2: 32 tokens per workgroup over all 200704 tokens
  size_t lds2 = (size_t)(32 * 200 + 32 * 776) * sizeof(float);
  int nblk2 = (16 * HH * HH) / 32;  // 6272
  k_mlp<<<nblk2, 256, lds2, stream>>>(g2, be2, w1, b1, w2, b2, out);
}